// conduitGNN_9852654977700
// MI455X (gfx1250) — compile-verified
//
#include <hip/hip_runtime.h>
#include <hip/hip_bf16.h>

typedef float v2f __attribute__((ext_vector_type(2)));
typedef float v8f __attribute__((ext_vector_type(8)));

__device__ __forceinline__ float sigmoidf(float x) {
    return 1.0f / (1.0f + __expf(-x));
}

// ---------------------------------------------------------------------------
// zero fill
// ---------------------------------------------------------------------------
__global__ __launch_bounds__(256) void gnn_zero_f(float* __restrict__ p, int n) {
    int i = blockIdx.x * blockDim.x + threadIdx.x;
    if (i < n) p[i] = 0.0f;
}

// ---------------------------------------------------------------------------
// agg[dst[e], :] += emb[src[e], :]   (float4 chunks, f32 atomics -> L2)
// ---------------------------------------------------------------------------
__global__ __launch_bounds__(256) void gnn_scatter_add(
    const float* __restrict__ emb, const int* __restrict__ src,
    const int* __restrict__ dst, float* __restrict__ agg,
    int nEdges, int d)
{
    const int chunks = d >> 2;
    long long idx = (long long)blockIdx.x * blockDim.x + threadIdx.x;
    long long total = (long long)nEdges * chunks;
    if (idx >= total) return;
    int e = (int)(idx / chunks);
    int c = (int)(idx % chunks) << 2;
    int s = src[e], t = dst[e];
    const float4 v = *(const float4*)(emb + (size_t)s * d + c);
    float* ap = agg + (size_t)t * d + c;
    atomicAdd(ap + 0, v.x);
    atomicAdd(ap + 1, v.y);
    atomicAdd(ap + 2, v.z);
    atomicAdd(ap + 3, v.w);
}

// ---------------------------------------------------------------------------
// out = sigmoid((P + Q) @ W + bias),  P,Q: [nRows,K] fp32, W: [K,Dout] row-major.
// One wave per 16-ROW tile; the wave computes ALL Dout/16 column tiles so each
// A fragment is loaded once and reused across colTiles WMMAs.
// V_WMMA_F32_16X16X4_F32 fragment layouts:
//   A 16x4: lanes 0-15 hold M=lane K={0,1}; lanes 16-31 hold M=lane-16 K={2,3}.
//   B 4x16: lanes 0-15 hold N=lane K={0,1}; lanes 16-31 hold N=lane-16 K={2,3}.
//   D:      vgpr r: lanes 0-15 -> (M=r, N=lane); lanes 16-31 -> (M=8+r, N=lane-16).
// ---------------------------------------------------------------------------
template <int K, int DOUT>
__global__ __launch_bounds__(256) void gnn_node_gemm_sigmoid(
    const float* __restrict__ P, const float* __restrict__ Q,
    const float* __restrict__ W, const float* __restrict__ bias,
    float* __restrict__ out, int nRows)
{
    constexpr int CT = DOUT / 16;
    const int lane = threadIdx.x & 31;
    const int wv   = threadIdx.x >> 5;
    const int rowTiles = (nRows + 15) >> 4;
    const int tileM = blockIdx.x * 8 + wv;
    if (tileM >= rowTiles) return;                    // wave-uniform

    const int rl   = lane & 15;
    const int hi   = lane >> 4;      // 0 or 1
    const int kofs = hi * 2;         // 0 or 2

    int row = tileM * 16 + rl;
    if (row >= nRows) row = nRows - 1;                // clamp (defensive)

    const float* prow = P + (size_t)row * K;
    const float* qrow = Q + (size_t)row * K;

    v8f acc[CT];
#pragma unroll
    for (int ct = 0; ct < CT; ++ct) acc[ct] = (v8f){};

    for (int kk = 0; kk < K; kk += 4) {
        const int k0 = kk + kofs;
        v2f a;
        a.x = prow[k0]     + qrow[k0];
        a.y = prow[k0 + 1] + qrow[k0 + 1];
        const float* wr0 = W + (size_t)k0 * DOUT + rl;
        const float* wr1 = wr0 + DOUT;
#pragma unroll
        for (int ct = 0; ct < CT; ++ct) {
            v2f b;
            b.x = wr0[ct * 16];
            b.y = wr1[ct * 16];
            acc[ct] = __builtin_amdgcn_wmma_f32_16x16x4_f32(false, a, false, b,
                                                            (short)0, acc[ct], false, false);
        }
    }

    const int mBase = tileM * 16 + hi * 8;
#pragma unroll
    for (int ct = 0; ct < CT; ++ct) {
        const int col = ct * 16 + rl;
        const float bcol = bias[col];
#pragma unroll
        for (int r = 0; r < 8; ++r) {
            const int m = mBase + r;
            if (m < nRows) out[(size_t)m * DOUT + col] = sigmoidf(acc[ct][r] + bcol);
        }
    }
}

// ---------------------------------------------------------------------------
// Fused edge pipeline: per 16-edge tile per wave:
//   c1  = sigmoid((h1[s]+h1[d]) @ Wc1 + bc1)          [16,64]  -> LDS
//   c2  = sigmoid((h2[s]+h2[d]) @ Wc2 + bc2)          [16,32]  -> D-frags
//   g1  = sigmoid(c1 @ Wg1 + bg1 + c2)                [16,32]  -> LDS
//   c3  = sigmoid((h3[s]+h3[d]) @ Wc3 + bc3)          [16,1]   (lanes 16-31)
//   out = sigmoid(g1 @ Wg2 + bg2 + c3)                [16,1]   (lanes 0-15)
// A fragments (gathered rows / LDS rows) loaded ONCE per K-step, reused
// across all column tiles. No [E,·] intermediates touch global memory.
// ---------------------------------------------------------------------------
#define C1_STRIDE 68   // 64 + pad (bank spread)
#define G1_STRIDE 36   // 32 + pad (rl*36 mod 64 distinct for rl=0..15)

__global__ __launch_bounds__(256) void gnn_edge_fused(
    const float* __restrict__ h1, const float* __restrict__ h2,
    const float* __restrict__ h3,
    const int* __restrict__ src, const int* __restrict__ dst,
    const float* __restrict__ Wc1, const float* __restrict__ bc1,
    const float* __restrict__ Wc2, const float* __restrict__ bc2,
    const float* __restrict__ Wg1, const float* __restrict__ bg1,
    const float* __restrict__ Wc3, const float* __restrict__ bc3,
    const float* __restrict__ Wg2, const float* __restrict__ bg2,
    float* __restrict__ out, int nEdges)
{
    __shared__ float ldsC1[8][16 * C1_STRIDE];
    __shared__ float ldsG1[8][16 * G1_STRIDE];

    const int lane = threadIdx.x & 31;
    const int wv   = threadIdx.x >> 5;
    const int tile = blockIdx.x * 8 + wv;
    const int nTiles = (nEdges + 15) >> 4;
    if (tile >= nTiles) return;                       // wave-uniform

    const int rl   = lane & 15;
    const int hi   = lane >> 4;
    const int kofs = hi * 2;

    int e = tile * 16 + rl;
    if (e >= nEdges) e = nEdges - 1;                  // clamp (defensive)
    const int sa = src[e], sb = dst[e];

    float* c1t = ldsC1[wv];
    float* g1t = ldsG1[wv];

    // ---- c1: K=128, Dout=64 (4 col tiles); gathered A loaded once per K-step ----
    {
        const float* pa = h1 + (size_t)sa * 128;
        const float* pb = h1 + (size_t)sb * 128;
        v8f acc[4];
#pragma unroll
        for (int ct = 0; ct < 4; ++ct) acc[ct] = (v8f){};
        for (int kk = 0; kk < 128; kk += 4) {
            const int k0 = kk + kofs;
            v2f a;
            a.x = pa[k0]     + pb[k0];
            a.y = pa[k0 + 1] + pb[k0 + 1];
            const float* wr0 = Wc1 + k0 * 64 + rl;
#pragma unroll
            for (int ct = 0; ct < 4; ++ct) {
                v2f b;
                b.x = wr0[ct * 16];
                b.y = wr0[ct * 16 + 64];
                acc[ct] = __builtin_amdgcn_wmma_f32_16x16x4_f32(false, a, false, b,
                                                                (short)0, acc[ct], false, false);
            }
        }
#pragma unroll
        for (int ct = 0; ct < 4; ++ct) {
            const int col = ct * 16 + rl;
            const float bcol = bc1[col];
#pragma unroll
            for (int r = 0; r < 8; ++r)
                c1t[(hi * 8 + r) * C1_STRIDE + col] = sigmoidf(acc[ct][r] + bcol);
        }
    }

    // ---- c2: K=64, Dout=32 (2 col tiles); keep sigmoid'd D-frags in regs ----
    v8f c2f[2];
    {
        const float* qa = h2 + (size_t)sa * 64;
        const float* qb = h2 + (size_t)sb * 64;
        v8f acc[2];
#pragma unroll
        for (int ct = 0; ct < 2; ++ct) acc[ct] = (v8f){};
        for (int kk = 0; kk < 64; kk += 4) {
            const int k0 = kk + kofs;
            v2f a;
            a.x = qa[k0]     + qb[k0];
            a.y = qa[k0 + 1] + qb[k0 + 1];
            const float* wr0 = Wc2 + k0 * 32 + rl;
#pragma unroll
            for (int ct = 0; ct < 2; ++ct) {
                v2f b;
                b.x = wr0[ct * 16];
                b.y = wr0[ct * 16 + 32];
                acc[ct] = __builtin_amdgcn_wmma_f32_16x16x4_f32(false, a, false, b,
                                                                (short)0, acc[ct], false, false);
            }
        }
#pragma unroll
        for (int ct = 0; ct < 2; ++ct) {
            const float bcol = bc2[ct * 16 + rl];
#pragma unroll
            for (int r = 0; r < 8; ++r) c2f[ct][r] = sigmoidf(acc[ct][r] + bcol);
        }
    }

    // ---- g1: K=64 (A from LDS c1), Dout=32; c2 D-frag layout matches -> add ----
    {
        v8f acc[2];
#pragma unroll
        for (int ct = 0; ct < 2; ++ct) acc[ct] = (v8f){};
        for (int kk = 0; kk < 64; kk += 4) {
            const int k0 = kk + kofs;
            v2f a;
            a.x = c1t[rl * C1_STRIDE + k0];
            a.y = c1t[rl * C1_STRIDE + k0 + 1];
            const float* wr0 = Wg1 + k0 * 32 + rl;
#pragma unroll
            for (int ct = 0; ct < 2; ++ct) {
                v2f b;
                b.x = wr0[ct * 16];
                b.y = wr0[ct * 16 + 32];
                acc[ct] = __builtin_amdgcn_wmma_f32_16x16x4_f32(false, a, false, b,
                                                                (short)0, acc[ct], false, false);
            }
        }
#pragma unroll
        for (int ct = 0; ct < 2; ++ct) {
            const int col = ct * 16 + rl;
            const float bcol = bg1[col];
#pragma unroll
            for (int r = 0; r < 8; ++r)
                g1t[(hi * 8 + r) * G1_STRIDE + col] = sigmoidf(acc[ct][r] + bcol + c2f[ct][r]);
        }
    }

    // ---- heads: lanes 0-15 do g1·Wg2 ; lanes 16-31 do c3 from h3 gather ----
    float partial;
    if (hi == 0) {
        float s = 0.0f;
#pragma unroll
        for (int j = 0; j < 32; ++j) s += g1t[rl * G1_STRIDE + j] * Wg2[j];
        partial = s;                                  // pre-activation g1·Wg2
    } else {
        const float* ra = h3 + (size_t)sa * 32;
        const float* rb = h3 + (size_t)sb * 32;
        float s = 0.0f;
#pragma unroll
        for (int j = 0; j < 32; ++j) s += (ra[j] + rb[j]) * Wc3[j];
        partial = sigmoidf(s + bc3[0]);               // c3
    }
    const float c3v = __shfl_xor(partial, 16, 32);    // lanes 0-15 fetch c3
    if (hi == 0) {
        const int eo = tile * 16 + rl;
        if (eo < nEdges) out[eo] = sigmoidf(partial + bg2[0] + c3v);
    }
}

// ---------------------------------------------------------------------------
extern "C" void kernel_launch(void* const* d_in, const int* in_sizes, int n_in,
                              void* d_out, int out_size, void* d_ws, size_t ws_size,
                              hipStream_t stream) {
    const float* x   = (const float*)d_in[0];
    const int*   src = (const int*)d_in[1];
    const int*   dst = (const int*)d_in[2];
    const float* Wn1 = (const float*)d_in[3],  *bn1 = (const float*)d_in[4];
    const float* Wc1 = (const float*)d_in[5],  *bc1 = (const float*)d_in[6];
    const float* Wn2 = (const float*)d_in[7],  *bn2 = (const float*)d_in[8];
    const float* Wc2 = (const float*)d_in[9],  *bc2 = (const float*)d_in[10];
    const float* Wn3 = (const float*)d_in[11], *bn3 = (const float*)d_in[12];
    const float* Wc3 = (const float*)d_in[13], *bc3 = (const float*)d_in[14];
    const float* Wg1 = (const float*)d_in[15], *bg1 = (const float*)d_in[16];
    const float* Wg2 = (const float*)d_in[17], *bg2 = (const float*)d_in[18];
    float* out = (float*)d_out;

    const int n  = in_sizes[0] / 64;   // 50000 nodes
    const int ne = in_sizes[1];        // 800000 edges

    // workspace: agg[n*128] (reused), h1[n*128], h2[n*64], h3[n*32]  (~70 MB)
    float* agg = (float*)d_ws;
    float* h1  = agg + (size_t)n * 128;
    float* h2  = h1  + (size_t)n * 128;
    float* h3  = h2  + (size_t)n * 64;

    auto blocks1d = [](long long work) { return (int)((work + 255) / 256); };
    const int rowTiles  = (n + 15) / 16;
    const int nodeBlks  = (rowTiles + 7) / 8;         // 8 waves/block, 1 row-tile/wave

    // ---- stage 1: h1 = sigmoid((x + agg(x)) @ Wn1 + bn1), K=64 -> 128 ----
    gnn_zero_f<<<blocks1d((long long)n * 64), 256, 0, stream>>>(agg, n * 64);
    gnn_scatter_add<<<blocks1d((long long)ne * 16), 256, 0, stream>>>(x, src, dst, agg, ne, 64);
    gnn_node_gemm_sigmoid<64, 128><<<nodeBlks, 256, 0, stream>>>(x, agg, Wn1, bn1, h1, n);

    // ---- stage 2: h2 = sigmoid((h1 + agg(h1)) @ Wn2 + bn2), K=128 -> 64 ----
    gnn_zero_f<<<blocks1d((long long)n * 128), 256, 0, stream>>>(agg, n * 128);
    gnn_scatter_add<<<blocks1d((long long)ne * 32), 256, 0, stream>>>(h1, src, dst, agg, ne, 128);
    gnn_node_gemm_sigmoid<128, 64><<<nodeBlks, 256, 0, stream>>>(h1, agg, Wn2, bn2, h2, n);

    // ---- stage 3: h3 = sigmoid((h2 + agg(h2)) @ Wn3 + bn3), K=64 -> 32 ----
    gnn_zero_f<<<blocks1d((long long)n * 64), 256, 0, stream>>>(agg, n * 64);
    gnn_scatter_add<<<blocks1d((long long)ne * 16), 256, 0, stream>>>(h2, src, dst, agg, ne, 64);
    gnn_node_gemm_sigmoid<64, 32><<<nodeBlks, 256, 0, stream>>>(h2, agg, Wn3, bn3, h3, n);

    // ---- fused edge pipeline: c1, c2, g1, c3, out — no [E,·] spills ----
    int edgeTiles = (ne + 15) / 16;
    gnn_edge_fused<<<(edgeTiles + 7) / 8, 256, 0, stream>>>(
        h1, h2, h3, src, dst,
        Wc1, bc1, Wc2, bc2, Wg1, bg1, Wc3, bc3, Wg2, bg2,
        out, ne);
}